// Attention_43542378447097
// MI455X (gfx1250) — compile-verified
//
#include <hip/hip_runtime.h>

typedef __attribute__((ext_vector_type(2))) float v2f;
typedef __attribute__((ext_vector_type(4))) float v4f;
typedef __attribute__((ext_vector_type(8))) float v8f;

// Problem constants
constexpr int Bn = 96, Cn = 96, Sn = 96, NHn = 8, DKn = 96, Dn = 768, Gn = 8;
constexpr float EPSf = 1e-5f;

#define WMMA_F32(a, b, c) \
    __builtin_amdgcn_wmma_f32_16x16x4_f32(false, (a), false, (b), (short)0, (c), false, false)

// ---------------------------------------------------------------------------
// GroupNorm: one block per (b, group). Group = 12 channels x 96 = 1152 elems.
// ---------------------------------------------------------------------------
__global__ void __launch_bounds__(128)
gn_kernel(const float* __restrict__ x, const float* __restrict__ gamma,
          const float* __restrict__ beta, float* __restrict__ xn) {
    const int bg = blockIdx.x;
    const int b = bg / Gn, g = bg % Gn;
    const int CPG = Cn / Gn;            // 12
    const int N = CPG * Cn;             // 1152
    const float* xp = x + (size_t)(b * Cn + g * CPG) * Cn;
    float s = 0.f, s2 = 0.f;
    for (int i = threadIdx.x; i < N; i += 128) {
        float v = xp[i];
        s += v; s2 += v * v;
    }
    __shared__ float rs[128], rs2[128];
    rs[threadIdx.x] = s; rs2[threadIdx.x] = s2;
    __syncthreads();
    for (int off = 64; off > 0; off >>= 1) {
        if (threadIdx.x < off) {
            rs[threadIdx.x]  += rs[threadIdx.x + off];
            rs2[threadIdx.x] += rs2[threadIdx.x + off];
        }
        __syncthreads();
    }
    const float mu  = rs[0] / (float)N;
    const float var = rs2[0] / (float)N - mu * mu;
    const float inv = rsqrtf(var + EPSf);
    float* op = xn + (size_t)(b * Cn + g * CPG) * Cn;
    for (int i = threadIdx.x; i < N; i += 128) {
        const int c = g * CPG + i / Cn;
        op[i] = (xp[i] - mu) * inv * gamma[c] + beta[c];
    }
}

// ---------------------------------------------------------------------------
// fp32 WMMA GEMM + bias: C[M,N] = A[M,K] @ B[K,N] + bias[N]
// 2x2 register-blocked: each wave owns a 32x32 macro-tile (4 accumulators),
// so each A/B fragment pair feeds two WMMAs -> 4 v_wmma per 6 VMEM ops.
// Requires M%32==0, N%32==0, K%4==0 (holds for all call sites).
// Frag layout (wave32, V_WMMA_F32_16X16X4_F32):
//   A: M = lane%16, K pair = {0,1} (lanes 0-15) / {2,3} (lanes 16-31)
//   B: N = lane%16, same K-pair split;  C/D: N = lane%16, M = i + 8*(lane/16)
// ---------------------------------------------------------------------------
__global__ void __launch_bounds__(256)
gemm_bias_wmma(const float* __restrict__ A, const float* __restrict__ Bm,
               const float* __restrict__ bias, float* __restrict__ Cm,
               int M, int N, int K) {
    const int lane = threadIdx.x & 31;
    const int wave = threadIdx.x >> 5;
    const int tilesN = N >> 5;                 // 32-wide macro tiles
    const int totalTiles = (M >> 5) * tilesN;
    const int tile = blockIdx.x * 8 + wave;    // wave-uniform
    if (tile >= totalTiles) return;
    const int tm = tile / tilesN, tn = tile % tilesN;
    const int half = lane >> 4, lm = lane & 15;

    const float* Arow0 = A + (size_t)(tm * 32 + lm) * K;
    const float* Arow1 = Arow0 + (size_t)16 * K;
    const float* Bcol0 = Bm + tn * 32 + lm;
    const float* Bcol1 = Bcol0 + 16;

    __builtin_prefetch(Arow0, 0, 1);
    __builtin_prefetch(Arow1, 0, 1);
    __builtin_prefetch(Bcol0, 0, 1);

    v8f c00 = (v8f)0.0f, c01 = (v8f)0.0f, c10 = (v8f)0.0f, c11 = (v8f)0.0f;
    for (int kk = 0; kk < K; kk += 4) {
        const int k0 = kk + half * 2;
        v2f a0, a1, b0, b1;
        a0.x = Arow0[k0];  a0.y = Arow0[k0 + 1];
        a1.x = Arow1[k0];  a1.y = Arow1[k0 + 1];
        b0.x = Bcol0[(size_t)k0 * N];  b0.y = Bcol0[(size_t)(k0 + 1) * N];
        b1.x = Bcol1[(size_t)k0 * N];  b1.y = Bcol1[(size_t)(k0 + 1) * N];
        c00 = WMMA_F32(a0, b0, c00);
        c01 = WMMA_F32(a0, b1, c01);
        c10 = WMMA_F32(a1, b0, c10);
        c11 = WMMA_F32(a1, b1, c11);
    }
    const int cn0 = tn * 32 + lm, cn1 = cn0 + 16;
    const float bv0 = bias[cn0], bv1 = bias[cn1];
#pragma unroll
    for (int i = 0; i < 8; ++i) {
        const int r0 = tm * 32 + half * 8 + i;
        const int r1 = r0 + 16;
        Cm[(size_t)r0 * N + cn0] = c00[i] + bv0;
        Cm[(size_t)r0 * N + cn1] = c01[i] + bv1;
        Cm[(size_t)r1 * N + cn0] = c10[i] + bv0;
        Cm[(size_t)r1 * N + cn1] = c11[i] + bv1;
    }
}

// ---------------------------------------------------------------------------
// Attention per (b, head): S = QK^T * scale, causal mask, softmax, O = P V.
// Scores (96x96, 36 KB) live in LDS so all 8 waves share the softmax.
// ---------------------------------------------------------------------------
__global__ void __launch_bounds__(256)
attn_kernel(const float* __restrict__ Qg, const float* __restrict__ Kg,
            const float* __restrict__ Vg, float* __restrict__ AOg) {
    __shared__ float PS[96 * 96];                    // scores / probabilities
    const int bh = blockIdx.x;
    const int b = bh >> 3, h = bh & 7;
    const size_t base = (size_t)b * Sn * Dn + (size_t)h * DKn;

    const int lane = threadIdx.x & 31;
    const int wave = threadIdx.x >> 5;
    const int half = lane >> 4, lm = lane & 15;
    const float scale = 0.10206207261596575f;        // 1/sqrt(96)

    // ---- scores = Q @ K^T (36 tiles of 16x16) ----
    for (int tile = wave; tile < 36; tile += 8) {    // wave-uniform loop
        const int tm = tile / 6, tn = tile % 6;
        const float* Arow = Qg + base + (size_t)(tm * 16 + lm) * Dn;
        const float* Brow = Kg + base + (size_t)(tn * 16 + lm) * Dn;
        v8f c = (v8f)0.0f;
        for (int kk = 0; kk < 96; kk += 4) {
            const int k0 = kk + half * 2;
            v2f a, bb;
            a.x  = Arow[k0];     a.y  = Arow[k0 + 1];   // Q[m][k]
            bb.x = Brow[k0];     bb.y = Brow[k0 + 1];   // (K^T)[k][n] = K[n][k]
            c = WMMA_F32(a, bb, c);
        }
        const int coln = tn * 16 + lm;
#pragma unroll
        for (int i = 0; i < 8; ++i) {
            const int row = tm * 16 + half * 8 + i;
            PS[row * 96 + coln] = (coln <= row) ? c[i] * scale : -3.0e38f;
        }
    }
    __syncthreads();

    // ---- causal softmax, one row per thread (rows 0..95) ----
    if (threadIdx.x < 96) {
        float* row = PS + threadIdx.x * 96;
        float m = -3.4e38f;
        for (int j = 0; j < 96; ++j) m = fmaxf(m, row[j]);
        float sum = 0.f;
        for (int j = 0; j < 96; ++j) { float e = __expf(row[j] - m); row[j] = e; sum += e; }
        const float r = 1.0f / sum;
        for (int j = 0; j < 96; ++j) row[j] *= r;
    }
    __syncthreads();

    // ---- out = P @ V ----
    for (int tile = wave; tile < 36; tile += 8) {
        const int tm = tile / 6, tn = tile % 6;
        const float* Arow = PS + (tm * 16 + lm) * 96;
        const float* Bcol = Vg + base + tn * 16 + lm;
        v8f c = (v8f)0.0f;
        for (int kk = 0; kk < 96; kk += 4) {
            const int k0 = kk + half * 2;
            v2f a, bb;
            a.x  = Arow[k0];                a.y  = Arow[k0 + 1];          // P (LDS)
            bb.x = Bcol[(size_t)k0 * Dn];   bb.y = Bcol[(size_t)(k0 + 1) * Dn];
            c = WMMA_F32(a, bb, c);
        }
        const int coln = tn * 16 + lm;
#pragma unroll
        for (int i = 0; i < 8; ++i) {
            const int row = tm * 16 + half * 8 + i;
            AOg[base + (size_t)row * Dn + coln] = c[i];
        }
    }
}

// ---------------------------------------------------------------------------
// Broadcast residual add: out[b,c,s,:] = O[b,s,:] + xn[b,c,:]
// 340 MB streaming write -> float4 + nontemporal stores (don't pollute L2).
// ---------------------------------------------------------------------------
__global__ void __launch_bounds__(256)
add_residual(const float* __restrict__ O, const float* __restrict__ xn,
             float* __restrict__ out) {
    const long long idx4 = (long long)blockIdx.x * 256 + threadIdx.x;
    const long long total4 = (long long)Bn * Cn * Sn * (Cn / 4);
    if (idx4 >= total4) return;
    const int q = (int)(idx4 % 24);
    long long r = idx4 / 24;
    const int s = (int)(r % 96); r /= 96;
    const int c = (int)(r % 96);
    const long long b = r / 96;
    const v4f o = *(const v4f*)(O  + ((b * 96 + s) * 96 + q * 4));
    const v4f x = *(const v4f*)(xn + ((b * 96 + c) * 96 + q * 4));
    __builtin_nontemporal_store(o + x, (v4f*)out + idx4);
}

// ---------------------------------------------------------------------------
extern "C" void kernel_launch(void* const* d_in, const int* in_sizes, int n_in,
                              void* d_out, int out_size, void* d_ws, size_t ws_size,
                              hipStream_t stream) {
    const float* x     = (const float*)d_in[0];
    const float* Wq    = (const float*)d_in[1];
    const float* bq    = (const float*)d_in[2];
    const float* Wk    = (const float*)d_in[3];
    const float* bk    = (const float*)d_in[4];
    const float* Wv    = (const float*)d_in[5];
    const float* bv    = (const float*)d_in[6];
    const float* Wo    = (const float*)d_in[7];
    const float* bo    = (const float*)d_in[8];
    const float* gamma = (const float*)d_in[9];
    const float* beta  = (const float*)d_in[10];
    float* out = (float*)d_out;

    // Workspace layout (floats)
    float* xn = (float*)d_ws;            // 96*96*96      =   884,736
    float* Q  = xn + 884736;             // 96*96*768     = 7,077,888
    float* K  = Q  + 7077888;
    float* V  = K  + 7077888;
    float* AO = V  + 7077888;
    float* O  = AO + 7077888;            // 96*96*96

    // 1) GroupNorm
    gn_kernel<<<Bn * Gn, 128, 0, stream>>>(x, gamma, beta, xn);

    // 2) QKV projections: M=9216, N=768, K=96 -> (288*24)=6912 macro-tiles / 8 waves
    gemm_bias_wmma<<<864, 256, 0, stream>>>(xn, Wq, bq, Q, 9216, 768, 96);
    gemm_bias_wmma<<<864, 256, 0, stream>>>(xn, Wk, bk, K, 9216, 768, 96);
    gemm_bias_wmma<<<864, 256, 0, stream>>>(xn, Wv, bv, V, 9216, 768, 96);

    // 3) Attention: one block per (batch, head)
    attn_kernel<<<Bn * NHn, 256, 0, stream>>>(Q, K, V, AO);

    // 4) Output projection: M=9216, N=96, K=768 -> (288*3)=864 macro-tiles
    gemm_bias_wmma<<<108, 256, 0, stream>>>(AO, Wo, bo, O, 9216, 96, 768);

    // 5) Broadcast residual add (340 MB streaming output)
    add_residual<<<82944, 256, 0, stream>>>(O, xn, out);
}